// MACENet_53807350284827
// MI455X (gfx1250) — compile-verified
//
#include <hip/hip_runtime.h>
#include <math.h>

// ---------------- CDNA5 WMMA (f32 16x16x4) ----------------
typedef float v2f __attribute__((ext_vector_type(2)));
typedef float v8f __attribute__((ext_vector_type(8)));

__device__ inline v8f wmma4(v2f a, v2f b, v8f c) {
  // D = A(16x4) * B(4x16) + C, f32 everywhere
  return __builtin_amdgcn_wmma_f32_16x16x4_f32(false, a, false, b, (short)0, c, false, false);
}

#define NP 34
#define NZCAP 2048

__constant__ int c_l1[NP] = {0,0,0,0, 1,1,1,1,1,1,1,1,1, 2,2,2,2,2,2,2,2,2,2,2, 3,3,3,3,3,3,3,3,3,3};
__constant__ int c_l2[NP] = {0,1,2,3, 0,1,1,1,2,2,2,3,3, 0,1,1,1,2,2,2,2,3,3,3, 0,1,1,2,2,2,3,3,3,3};
__constant__ int c_l3[NP] = {0,1,2,3, 1,0,1,2,1,2,3,2,3, 2,1,2,3,0,1,2,3,1,2,3, 3,2,3,1,2,3,0,1,2,3};
__constant__ int c_ltab[16] = {0,1,1,1,2,2,2,2,2,3,3,3,3,3,3,3};
__constant__ float c_invs[4] = {1.0f, 0.57735026919f, 0.44721359550f, 0.37796447301f};

__device__ inline float silu_f(float x) { return x / (1.0f + expf(-x)); }

__device__ inline void atomicAddF(float* p, float v) {
  unsafeAtomicAdd(p, v);   // GLOBAL_ATOMIC_ADD_F32
}

// ---------------- on-device Wigner-3j construction ----------------
__device__ double fct(int n) {
  const double F[11] = {1.,1.,2.,6.,24.,120.,720.,5040.,40320.,362880.,3628800.};
  return F[n];
}

__device__ double su2cg(int j1, int j2, int j3, int m1, int m2) {
  int m3 = m1 + m2;
  double pref = (2.0*j3+1.0)*fct(j3+j1-j2)*fct(j3-j1+j2)*fct(j1+j2-j3)/fct(j1+j2+j3+1);
  pref *= fct(j3+m3)*fct(j3-m3)*fct(j1-m1)*fct(j1+m1)*fct(j2-m2)*fct(j2+m2);
  pref = sqrt(pref);
  int k0 = 0; if (j2-j3-m1 > k0) k0 = j2-j3-m1; if (j1-j3+m2 > k0) k0 = j1-j3+m2;
  int k1 = j1+j2-j3; if (j1-m1 < k1) k1 = j1-m1; if (j2+m2 < k1) k1 = j2+m2;
  double s = 0.0;
  for (int k = k0; k <= k1; ++k) {
    double d = fct(k)*fct(j1+j2-j3-k)*fct(j1-m1-k)*fct(j2+m2-k)*fct(j3-j2+m1+k)*fct(j3-j1-m2+k);
    s += ((k & 1) ? -1.0 : 1.0) / d;
  }
  return pref * s;
}

__device__ void qval(int l, int row, int col, double* re, double* im) {
  // e3nn complex<-real change of basis, times (-i)^l
  int m = row - l;
  double vr = 0.0, vi = 0.0;
  const double is2 = 0.70710678118654752440;
  if (m < 0) {
    if (col == l - m)      { vr = is2; vi = 0.0; }       // l+|m|
    else if (col == l + m) { vr = 0.0; vi = -is2; }      // l-|m|
  } else if (m == 0) {
    if (col == l) { vr = 1.0; vi = 0.0; }
  } else {
    double s = (m & 1) ? -1.0 : 1.0;
    if (col == l + m)      { vr = s*is2; vi = 0.0; }
    else if (col == l - m) { vr = 0.0;   vi = s*is2; }
  }
  switch (l & 3) {
    case 0: *re =  vr; *im =  vi; break;
    case 1: *re =  vi; *im = -vr; break;
    case 2: *re = -vr; *im = -vi; break;
    default:*re = -vi; *im =  vr; break;
  }
}

// one block per CG path; dense w3j[p][7][7][7]
__global__ void k_setup(float* __restrict__ w3j) {
  __shared__ double sCsu[343];
  __shared__ double sQ[3][49][2];
  int p = blockIdx.x, t = threadIdx.x;
  int j1 = c_l1[p], j2 = c_l2[p], j3 = c_l3[p];
  if (t < 147) {
    int which = t / 49, idx = t % 49, i = idx / 7, j = idx % 7;
    int l = (which == 0) ? j1 : (which == 1) ? j2 : j3;
    double re = 0.0, im = 0.0;
    if (i < 2*l+1 && j < 2*l+1) qval(l, i, j, &re, &im);
    sQ[which][idx][0] = re; sQ[which][idx][1] = im;
  }
  if (t < 343) {
    int a = t / 49, b = (t / 7) % 7, c = t % 7;
    double v = 0.0;
    if (a < 2*j1+1 && b < 2*j2+1 && c < 2*j3+1) {
      int m1 = a - j1, m2 = b - j2, m3 = c - j3;
      if (m1 + m2 == m3) v = su2cg(j1, j2, j3, m1, m2);
    }
    sCsu[t] = v;
  }
  __syncthreads();
  if (t < 343) {
    int i = t / 49, j = (t / 7) % 7, k = t % 7;
    double re = 0.0;
    if (i < 2*j1+1 && j < 2*j2+1 && k < 2*j3+1) {
      for (int a = 0; a < 2*j1+1; ++a)
        for (int b = 0; b < 2*j2+1; ++b)
          for (int c = 0; c < 2*j3+1; ++c) {
            double cv = sCsu[a*49 + b*7 + c];
            if (cv == 0.0) continue;
            double ar = sQ[0][a*7+i][0], ai = sQ[0][a*7+i][1];
            double br = sQ[1][b*7+j][0], bi = sQ[1][b*7+j][1];
            double cr = sQ[2][c*7+k][0], ci = -sQ[2][c*7+k][1];
            double xr = ar*br - ai*bi, xi = ar*bi + ai*br;
            re += cv * (xr*cr - xi*ci);
          }
    }
    w3j[p*343 + t] = (float)re;
  }
}

// single-thread: sparse (path,c)-sorted table + SH recursion scales
__global__ void k_build_sparse(const float* __restrict__ w3j, int* __restrict__ nzabc,
                               float* __restrict__ nzval, int* __restrict__ offpc,
                               float* __restrict__ shsc) {
  if (threadIdx.x != 0 || blockIdx.x != 0) return;
  int cnt = 0; offpc[0] = 0;
  for (int p = 0; p < NP; ++p) {
    int l1 = c_l1[p], l2 = c_l2[p], l3 = c_l3[p];
    for (int c = 0; c < 7; ++c) {
      if (c <= 2*l3) {
        for (int a = 0; a <= 2*l1; ++a)
          for (int b = 0; b <= 2*l2; ++b) {
            float v = w3j[p*343 + a*49 + b*7 + c];
            if (fabsf(v) > 1e-6f && cnt < NZCAP) { nzabc[cnt] = (a << 3) | b; nzval[cnt] = v; ++cnt; }
          }
      }
      offpc[p*7 + c + 1] = cnt;
    }
  }
  // SH scales (paths (1,1,2)=7 and (2,1,3)=16)
  double vx = 0.2, vy = -0.6, vz = 0.7;
  double nv = sqrt(vx*vx + vy*vy + vz*vz); vx /= nv; vy /= nv; vz /= nv;
  double s3 = sqrt(3.0);
  double y1[3] = {s3*vx, s3*vy, s3*vz};
  const float* wA = w3j + 7*343;
  double t2[5], n2 = 0.0;
  for (int c = 0; c < 5; ++c) {
    double s = 0.0;
    for (int a = 0; a < 3; ++a) for (int b = 0; b < 3; ++b)
      s += (double)wA[a*49 + b*7 + c] * y1[a] * y1[b];
    t2[c] = s; n2 += s*s;
  }
  double sc2 = sqrt(5.0) / sqrt(n2);
  double y2[5]; for (int c = 0; c < 5; ++c) y2[c] = sc2 * t2[c];
  const float* wB = w3j + 16*343;
  double t3[7], n3 = 0.0;
  for (int c = 0; c < 7; ++c) {
    double s = 0.0;
    for (int a = 0; a < 5; ++a) for (int b = 0; b < 3; ++b)
      s += (double)wB[a*49 + b*7 + c] * y2[a] * y1[b];
    t3[c] = s; n3 += s*s;
  }
  double sc3 = sqrt(7.0) / sqrt(n3);
  shsc[0] = (float)sc2; shsc[1] = (float)sc3;
}

// ---------------- per-edge geometry: r, Y[16], rad[8] ----------------
__global__ void k_edge_geom(const float* __restrict__ vec, const float* __restrict__ w3j,
                            const float* __restrict__ shsc, float* __restrict__ Y,
                            float* __restrict__ RAD, int E) {
  __shared__ float s112[45], s213[105];
  int t = threadIdx.x;
  if (t < 45)  s112[t] = w3j[7*343  + (t/15)*49 + ((t/5)%3)*7 + (t%5)];
  if (t < 105) s213[t] = w3j[16*343 + (t/21)*49 + ((t/7)%3)*7 + (t%7)];
  __syncthreads();
  int e = blockIdx.x * blockDim.x + t;
  if (e >= E) return;
  float vx = vec[e*3+0], vy = vec[e*3+1], vz = vec[e*3+2];
  float rr = sqrtf(vx*vx + vy*vy + vz*vz);
  float inv = 1.0f / fmaxf(rr, 1e-9f);
  float sc2 = shsc[0], sc3 = shsc[1];
  float y1[3] = {1.7320508075688772f*vx*inv, 1.7320508075688772f*vy*inv, 1.7320508075688772f*vz*inv};
  float* yo = Y + (size_t)e*16;
  yo[0] = 1.0f; yo[1] = y1[0]; yo[2] = y1[1]; yo[3] = y1[2];
  float y2[5];
  for (int c = 0; c < 5; ++c) {
    float s = 0.0f;
    for (int a = 0; a < 3; ++a) for (int b = 0; b < 3; ++b)
      s += s112[a*15 + b*5 + c] * y1[a] * y1[b];
    y2[c] = sc2 * s; yo[4+c] = y2[c];
  }
  for (int c = 0; c < 7; ++c) {
    float s = 0.0f;
    for (int a = 0; a < 5; ++a) for (int b = 0; b < 3; ++b)
      s += s213[a*21 + b*7 + c] * y2[a] * y1[b];
    yo[9+c] = sc3 * s;
  }
  float x = rr * 0.2f;                       // r / R_MAX
  float env = 0.0f;
  if (x < 1.0f) {
    float x5 = x*x*x*x*x;
    env = 1.0f - 21.0f*x5 + 35.0f*x5*x - 15.0f*x5*x*x;
  }
  float* ro = RAD + (size_t)e*8;
  for (int n = 1; n <= 8; ++n)
    ro[n-1] = 0.6324555320336759f * sinf(3.14159265358979f * x * (float)n) * inv * env;
}

// ---------------- feats init ----------------
__global__ void k_init_feats(const float* __restrict__ embed, const int* __restrict__ z,
                             float* __restrict__ A, int N) {
  int pair = blockIdx.x * blockDim.x + threadIdx.x;
  int n = pair >> 5, f = pair & 31;
  if (n >= N) return;
  float* dst = A + (size_t)n*512 + f*16;
  dst[0] = embed[z[n]*32 + f];
  for (int d = 1; d < 16; ++d) dst[d] = 0.0f;
}

// ---------------- radial MLP: h2 = silu(silu(rad@w0)@w1), WMMA ----------------
__global__ void __launch_bounds__(128) k_radial(const float* __restrict__ rad,
                                                const float* __restrict__ w0,
                                                const float* __restrict__ w1,
                                                float* __restrict__ h2out) {
  __shared__ float sW0[512];
  __shared__ float sW1[4096];
  __shared__ float sRad[128];
  __shared__ float sH1[1024];
  int t = threadIdx.x, blk = blockIdx.x;
  for (int i = t; i < 512; i += 128) sW0[i] = w0[i];
  for (int i = t; i < 4096; i += 128) sW1[i] = w1[i];
  sRad[t] = rad[(size_t)blk*128 + t];
  __syncthreads();
  int lane = t & 31, ct = t >> 5;
  int lo = lane & 15, hi = lane >> 4, klo = hi*2;
  v8f c = {0.0f};
  #pragma unroll
  for (int k = 0; k < 8; k += 4) {
    int kk = k + klo;
    v2f av; av.x = sRad[lo*8 + kk];         av.y = sRad[lo*8 + kk + 1];
    v2f bv; bv.x = sW0[kk*64 + ct*16 + lo]; bv.y = sW0[(kk+1)*64 + ct*16 + lo];
    c = wmma4(av, bv, c);
  }
  #pragma unroll
  for (int i = 0; i < 8; ++i) sH1[(i + hi*8)*64 + ct*16 + lo] = silu_f(c[i]);
  __syncthreads();
  v8f c2 = {0.0f};
  #pragma unroll
  for (int k = 0; k < 64; k += 4) {
    int kk = k + klo;
    v2f av; av.x = sH1[lo*64 + kk];         av.y = sH1[lo*64 + kk + 1];
    v2f bv; bv.x = sW1[kk*64 + ct*16 + lo]; bv.y = sW1[(kk+1)*64 + ct*16 + lo];
    c2 = wmma4(av, bv, c2);
  }
  #pragma unroll
  for (int i = 0; i < 8; ++i)
    h2out[((size_t)blk*16 + i + hi*8)*64 + ct*16 + lo] = silu_f(c2[i]);
}

// ---------------- per-l F x F linear (e3nn Linear), WMMA ----------------
__global__ void __launch_bounds__(32) k_lin(const float* __restrict__ x,
                                            const float* __restrict__ w,
                                            float* __restrict__ out, int N) {
  int lane = threadIdx.x;
  int lo = lane & 15, hi = lane >> 4, klo = hi*2;
  int rem = blockIdx.x, l, nt = N >> 4;
  for (l = 0; l < 4; ++l) { int tl = nt*(2*l+1); if (rem < tl) break; rem -= tl; }
  int tile = rem;
  int w21 = 2*l + 1, ll = l*l;
  int m = tile*16 + lo;
  int n = m / w21, aa = m - n*w21;
  const float* ax = x + (size_t)n*512 + ll + aa;
  const float* wl = w + l*1024;
  v8f c0 = {0.0f}, c1 = {0.0f};
  #pragma unroll
  for (int k = 0; k < 32; k += 4) {
    int kk = k + klo;
    v2f av;  av.x  = ax[kk*16];            av.y  = ax[(kk+1)*16];
    v2f b0v; b0v.x = wl[kk*32 + lo];       b0v.y = wl[(kk+1)*32 + lo];
    v2f b1v; b1v.x = wl[kk*32 + 16 + lo];  b1v.y = wl[(kk+1)*32 + 16 + lo];
    c0 = wmma4(av, b0v, c0);
    c1 = wmma4(av, b1v, c1);
  }
  #pragma unroll
  for (int i = 0; i < 8; ++i) {
    int m2 = tile*16 + i + hi*8;
    int n2 = m2 / w21, a2 = m2 - n2*w21;
    out[(size_t)n2*512 + lo*16 + ll + a2]        = c0[i];
    out[(size_t)n2*512 + (16 + lo)*16 + ll + a2] = c1[i];
  }
}

// ---------------- edge tensor product (fused rw GEMM per path) ----------------
__global__ void __launch_bounds__(256) k_tp_edge(const float* __restrict__ hs_in,
                                                 const float* __restrict__ h2,
                                                 const float* __restrict__ Y,
                                                 const float* __restrict__ w2,
                                                 const int* __restrict__ snd,
                                                 const int* __restrict__ rcv,
                                                 const int* __restrict__ g_abc,
                                                 const float* __restrict__ g_val,
                                                 const int* __restrict__ g_off,
                                                 float* __restrict__ agg, int E) {
  __shared__ float shY[128];          // 8 edges x 16
  __shared__ float shH2[1024];        // 16 x 64 (rows 8..15 zero pad for WMMA)
  __shared__ float shRW[256];         // 8 x 32
  __shared__ float shHS[4352];        // 8 x 32 x 17 (padded)
  __shared__ float shACC[4096];       // 16 dims x 256 threads
  __shared__ int   shABC[NZCAP];
  __shared__ float shVAL[NZCAP];
  __shared__ int   shOFF[240];
  int t = threadIdx.x, blk = blockIdx.x;
  int e_loc = t >> 5, f = t & 31;
  int e_glob = blk*8 + e_loc;

  for (int i = t; i < 240; i += 256) shOFF[i] = g_off[i];
  for (int i = t; i < NZCAP; i += 256) { shABC[i] = g_abc[i]; shVAL[i] = g_val[i]; }
  if (t < 128) shY[t] = Y[((size_t)blk*8 + (t >> 4))*16 + (t & 15)];
  for (int i = t; i < 1024; i += 256) {
    int row = i >> 6, col = i & 63;
    shH2[i] = (row < 8) ? h2[((size_t)blk*8 + row)*64 + col] : 0.0f;
  }
  {
    int s = snd[e_glob];
    const float4* src = (const float4*)(hs_in + (size_t)s*512 + f*16);
    float* dst = &shHS[e_loc*544 + f*17];
    #pragma unroll
    for (int q = 0; q < 4; ++q) {
      float4 v = src[q];
      dst[q*4+0] = v.x; dst[q*4+1] = v.y; dst[q*4+2] = v.z; dst[q*4+3] = v.w;
    }
  }
  for (int d = 0; d < 16; ++d) shACC[d*256 + t] = 0.0f;
  __syncthreads();

  int lane = t & 31;
  int lo = lane & 15, hi = lane >> 4, klo = hi*2;

  for (int p = 0; p < NP; ++p) {
    if (t < 64) {                     // waves 0-1: rw = h2 @ w2[:, p*32 : p*32+32]
      int ct = t >> 5;
      v8f cf = {0.0f};
      #pragma unroll
      for (int k = 0; k < 64; k += 4) {
        int kk = k + klo;
        v2f av; av.x = shH2[lo*64 + kk]; av.y = shH2[lo*64 + kk + 1];
        const float* wcol = w2 + (size_t)kk*1088 + p*32 + ct*16 + lo;
        v2f bv; bv.x = wcol[0]; bv.y = wcol[1088];
        cf = wmma4(av, bv, cf);
      }
      if (hi == 0) {
        #pragma unroll
        for (int i = 0; i < 8; ++i) shRW[i*32 + ct*16 + lo] = cf[i];
      }
    }
    __syncthreads();
    int l1 = c_l1[p], l3 = c_l3[p];
    int a0 = l1*l1, b0 = c_l2[p]*c_l2[p], c0 = l3*l3;
    float rw = shRW[e_loc*32 + f];
    const float* hsrow = &shHS[e_loc*544 + f*17];
    const float* yrow  = &shY[e_loc*16 + b0];
    for (int c = 0; c <= 2*l3; ++c) {
      int beg = shOFF[p*7 + c], end = shOFF[p*7 + c + 1];
      float tv = 0.0f;
      for (int idx = beg; idx < end; ++idx) {
        int ab = shABC[idx];
        tv += shVAL[idx] * hsrow[a0 + (ab >> 3)] * yrow[ab & 7];
      }
      shACC[(c0 + c)*256 + t] += rw * tv;
    }
    __syncthreads();
  }
  int r = rcv[e_glob];
  float* dst = agg + (size_t)r*512 + f*16;
  const float inv = 1.0f / 16.0f;     // 1 / AVG_NEIGH
  for (int d = 0; d < 16; ++d) atomicAddF(dst + d, shACC[d*256 + t] * inv);
}

// ---------------- per-species selector (layer 0) ----------------
__global__ void __launch_bounds__(32) k_selector(const float* __restrict__ xin,
                                                 const float* __restrict__ sel,
                                                 const int* __restrict__ z,
                                                 float* __restrict__ out) {
  __shared__ float sX[512];
  int n = blockIdx.x, g = threadIdx.x;
  const float* src = xin + (size_t)n*512;
  for (int i = g; i < 512; i += 32) sX[i] = src[i];
  __syncthreads();
  const float* selz = sel + (size_t)z[n]*4096;
  float res[16];
  for (int d = 0; d < 16; ++d) {
    const float* wl = selz + c_ltab[d]*1024 + g;
    float acc = 0.0f;
    for (int fq = 0; fq < 32; ++fq) acc += sX[fq*16 + d] * wl[fq*32];
    res[d] = acc;
  }
  float* dst = out + (size_t)n*512 + g*16;
  for (int d = 0; d < 16; ++d) dst[d] = res[d];
}

// ---------------- product basis: A, B2=tp(A,A), B3=tp(B2,A), combine ----------------
__global__ void __launch_bounds__(128) k_product(const float* __restrict__ xin,
                                                 const float* __restrict__ pb,
                                                 const int* __restrict__ z,
                                                 const int* __restrict__ g_abc,
                                                 const float* __restrict__ g_val,
                                                 const int* __restrict__ g_off,
                                                 float* __restrict__ out, int N) {
  __shared__ float sX[2048], sB2[2048], sB3[2048];  // [d][t], t<128
  __shared__ int   sABC[NZCAP];
  __shared__ float sVAL[NZCAP];
  __shared__ int   sOFF[240];
  int t = threadIdx.x;
  for (int i = t; i < 240; i += 128) sOFF[i] = g_off[i];
  for (int i = t; i < NZCAP; i += 128) { sABC[i] = g_abc[i]; sVAL[i] = g_val[i]; }
  int pair = blockIdx.x*128 + t;
  int n = pair >> 5, f = pair & 31;
  bool ok = (n < N);
  if (ok) {
    const float* src = xin + (size_t)n*512 + f*16;
    for (int d = 0; d < 16; ++d) { sX[d*128+t] = src[d]; sB2[d*128+t] = 0.0f; sB3[d*128+t] = 0.0f; }
  } else {
    for (int d = 0; d < 16; ++d) { sX[d*128+t] = 0.0f; sB2[d*128+t] = 0.0f; sB3[d*128+t] = 0.0f; }
  }
  __syncthreads();
  for (int p = 0; p < NP; ++p) {
    int a0 = c_l1[p]*c_l1[p], b0 = c_l2[p]*c_l2[p];
    int l3 = c_l3[p], c0 = l3*l3;
    float sc = c_invs[l3];
    for (int c = 0; c <= 2*l3; ++c) {
      int beg = sOFF[p*7 + c], end = sOFF[p*7 + c + 1];
      float tv = 0.0f;
      for (int idx = beg; idx < end; ++idx) {
        int ab = sABC[idx];
        tv += sVAL[idx] * sX[(a0 + (ab >> 3))*128 + t] * sX[(b0 + (ab & 7))*128 + t];
      }
      sB2[(c0 + c)*128 + t] += sc * tv;
    }
  }
  for (int p = 0; p < NP; ++p) {
    int a0 = c_l1[p]*c_l1[p], b0 = c_l2[p]*c_l2[p];
    int l3 = c_l3[p], c0 = l3*l3;
    float sc = c_invs[l3];
    for (int c = 0; c <= 2*l3; ++c) {
      int beg = sOFF[p*7 + c], end = sOFF[p*7 + c + 1];
      float tv = 0.0f;
      for (int idx = beg; idx < end; ++idx) {
        int ab = sABC[idx];
        tv += sVAL[idx] * sB2[(a0 + (ab >> 3))*128 + t] * sX[(b0 + (ab & 7))*128 + t];
      }
      sB3[(c0 + c)*128 + t] += sc * tv;
    }
  }
  if (ok) {
    int zz = z[n];
    float w0 = pb[zz*96 + f], w1 = pb[zz*96 + 32 + f], w2 = pb[zz*96 + 64 + f];
    float* dst = out + (size_t)n*512 + f*16;
    for (int d = 0; d < 16; ++d)
      dst[d] = w0*sX[d*128+t] + w1*sB2[d*128+t] + w2*sB3[d*128+t];
  }
}

// ---------------- readouts ----------------
__global__ void k_readout0(const float* __restrict__ A, const float* __restrict__ ro0,
                           float* __restrict__ e0b, int N) {
  int n = blockIdx.x * blockDim.x + threadIdx.x;
  if (n >= N) return;
  float acc = 0.0f;
  for (int f = 0; f < 32; ++f) acc += A[(size_t)n*512 + f*16] * ro0[f];
  e0b[n] = acc;
}

__global__ void __launch_bounds__(32) k_resid(const float* __restrict__ feats,
                                              const float* __restrict__ res1,
                                              const int* __restrict__ z,
                                              float* __restrict__ out) {
  __shared__ float sS[32];
  int n = blockIdx.x, g = threadIdx.x;
  sS[g] = feats[(size_t)n*512 + g*16];
  __syncthreads();
  const float* w = res1 + (size_t)z[n]*1024 + g;
  float acc = 0.0f;
  for (int fq = 0; fq < 32; ++fq) acc += sS[fq] * w[fq*32];
  out[n*32 + g] = acc;
}

__global__ void k_final(const float* __restrict__ Cb, const float* __restrict__ resid,
                        const float* __restrict__ e0b, const float* __restrict__ ro1a,
                        const float* __restrict__ ro1b, float* __restrict__ out, int N) {
  int n = blockIdx.x * blockDim.x + threadIdx.x;
  if (n >= N) return;
  float sc[32];
  for (int g = 0; g < 32; ++g) sc[g] = Cb[(size_t)n*512 + g*16] + resid[n*32 + g];
  float e1 = 0.0f;
  for (int h = 0; h < 16; ++h) {
    float u = 0.0f;
    for (int g = 0; g < 32; ++g) u += sc[g] * ro1a[g*16 + h];
    e1 += silu_f(u) * ro1b[h];
  }
  float e = -1.5f + 2.0f * (e0b[n] + e1) - 10.0f;   // MEAN + STD*(e0+e1) + E0
  atomicAddF(out, e);
}

// ---------------- host orchestration ----------------
extern "C" void kernel_launch(void* const* d_in, const int* in_sizes, int n_in,
                              void* d_out, int out_size, void* d_ws, size_t ws_size,
                              hipStream_t stream) {
  const int E = in_sizes[0] / 3;
  const int N = in_sizes[1];
  const float* vectors   = (const float*)d_in[0];
  const int*   node_z    = (const int*)d_in[1];
  const int*   senders   = (const int*)d_in[2];
  const int*   receivers = (const int*)d_in[3];
  const float* embed     = (const float*)d_in[4];
  const float* rw0[2] = {(const float*)d_in[5],  (const float*)d_in[12]};
  const float* rw1[2] = {(const float*)d_in[6],  (const float*)d_in[13]};
  const float* rw2[2] = {(const float*)d_in[7],  (const float*)d_in[14]};
  const float* up[2]  = {(const float*)d_in[8],  (const float*)d_in[15]};
  const float* post[2]= {(const float*)d_in[9],  (const float*)d_in[16]};
  const float* pb[2]  = {(const float*)d_in[10], (const float*)d_in[17]};
  const float* pblin[2]={(const float*)d_in[11], (const float*)d_in[18]};
  const float* sel0  = (const float*)d_in[19];
  const float* ro0   = (const float*)d_in[20];
  const float* res1  = (const float*)d_in[21];
  const float* ro1_a = (const float*)d_in[22];
  const float* ro1_b = (const float*)d_in[23];

  size_t off = 0;
  auto take = [&](size_t bytes) -> void* {
    void* p = (char*)d_ws + off;
    off += (bytes + 255) & ~(size_t)255;
    return p;
  };
  float* W3J   = (float*)take((size_t)NP*343*4);
  int*   NZABC = (int*)  take((size_t)NZCAP*4);
  float* NZVAL = (float*)take((size_t)NZCAP*4);
  int*   OFFPC = (int*)  take(256*4);
  float* SHSC  = (float*)take(64*4);
  float* Yb    = (float*)take((size_t)E*16*4);
  float* RADb  = (float*)take((size_t)E*8*4);
  float* H2b   = (float*)take((size_t)E*64*4);
  float* Ab    = (float*)take((size_t)N*512*4);
  float* Bb    = (float*)take((size_t)N*512*4);
  float* Cb    = (float*)take((size_t)N*512*4);
  float* Db    = (float*)take((size_t)N*512*4);
  float* E0b   = (float*)take((size_t)N*4);
  float* RESb  = (float*)take((size_t)N*32*4);
  (void)ws_size; (void)n_in; (void)out_size;

  // constants + geometry
  k_setup<<<NP, 352, 0, stream>>>(W3J);
  k_build_sparse<<<1, 1, 0, stream>>>(W3J, NZABC, NZVAL, OFFPC, SHSC);
  k_edge_geom<<<(E + 255)/256, 256, 0, stream>>>(vectors, W3J, SHSC, Yb, RADb, E);
  k_init_feats<<<(N*32 + 255)/256, 256, 0, stream>>>(embed, node_z, Ab, N);
  hipMemsetAsync(d_out, 0, sizeof(float), stream);

  for (int L = 0; L < 2; ++L) {
    k_radial<<<E/16, 128, 0, stream>>>(RADb, rw0[L], rw1[L], H2b);
    k_lin<<<N, 32, 0, stream>>>(Ab, up[L], Bb, N);
    hipMemsetAsync(Cb, 0, (size_t)N*512*4, stream);
    k_tp_edge<<<E/8, 256, 0, stream>>>(Bb, H2b, Yb, rw2[L], senders, receivers,
                                       NZABC, NZVAL, OFFPC, Cb, E);
    k_lin<<<N, 32, 0, stream>>>(Cb, post[L], Db, N);
    if (L == 0) {
      k_selector<<<N, 32, 0, stream>>>(Db, sel0, node_z, Bb);
      k_product<<<(N*32 + 127)/128, 128, 0, stream>>>(Bb, pb[0], node_z,
                                                      NZABC, NZVAL, OFFPC, Cb, N);
      k_lin<<<N, 32, 0, stream>>>(Cb, pblin[0], Ab, N);   // new feats
      k_readout0<<<(N + 255)/256, 256, 0, stream>>>(Ab, ro0, E0b, N);
      k_resid<<<N, 32, 0, stream>>>(Ab, res1, node_z, RESb);
    } else {
      k_product<<<(N*32 + 127)/128, 128, 0, stream>>>(Db, pb[1], node_z,
                                                      NZABC, NZVAL, OFFPC, Bb, N);
      k_lin<<<N, 32, 0, stream>>>(Bb, pblin[1], Cb, N);
      k_final<<<(N + 255)/256, 256, 0, stream>>>(Cb, RESb, E0b, ro1_a, ro1_b,
                                                 (float*)d_out, N);
    }
  }
}